// evalYolov2_89438398972629
// MI455X (gfx1250) — compile-verified
//
#include <hip/hip_runtime.h>
#include <cstdint>

// ---------------- problem constants (from reference) ----------------
constexpr int kClasses = 80;
constexpr int kNum     = 5;
constexpr int kAL      = kClasses + 4 + 1;   // 85
constexpr int kH       = 19;
constexpr int kW       = 19;
constexpr int kHW      = kH * kW;            // 361
constexpr int kB       = 32;
constexpr int kN       = kNum * kHW;         // 1805
constexpr int kNPad    = 2048;               // next pow2 for bitonic sort
constexpr float kObjTh = 0.5f;
constexpr float kNmsTh = 0.45f;

// ---------------- gfx1250 helpers ----------------
// LDS byte address for async-to-LDS instructions: flat shared addresses carry
// the wave-relative LDS offset in the low 32 bits.
static __device__ __forceinline__ unsigned ldsAddr(const void* p) {
  return (unsigned)(uintptr_t)p;
}

// GVS-form async copy: one dword from saddr + vaddr into LDS at vdst; tracked by ASYNCcnt.
static __device__ __forceinline__ void asyncLoadB32(unsigned ldsByteAddr,
                                                    unsigned byteOff,
                                                    unsigned long long sbase) {
  asm volatile("global_load_async_to_lds_b32 %0, %1, %2"
               :: "v"(ldsByteAddr), "v"(byteOff), "s"(sbase)
               : "memory");
}

static __device__ __forceinline__ void waitAsync0() {
  asm volatile("s_wait_asynccnt 0" ::: "memory");
}

// ---------------- kernel 1: decode ----------------
// det[b*kN + (a*kHW + cell)] row = [obj, x, y, w, h, cls, prob]
__global__ void yolo_decode_kernel(const float* __restrict__ pred,
                                   const float* __restrict__ anchors,
                                   float* __restrict__ det) {
  int t = blockIdx.x * blockDim.x + threadIdx.x;
  if (t >= kB * kN) return;
  int b    = t / kN;
  int rem  = t - b * kN;            // n = a*kHW + cell
  int a    = rem / kHW;
  int cell = rem - a * kHW;
  int jj   = cell / kW;
  int ii   = cell - jj * kW;

  const float* base = pred + (size_t)(b * (kNum * kAL) + a * kAL) * kHW + cell;
  // gfx1250 prefetch of the class-logit region (global_prefetch_b8)
  __builtin_prefetch(base + 5 * kHW, 0, 0);

  float r0  = base[0 * kHW];
  float r1  = base[1 * kHW];
  float r2  = base[2 * kHW];
  float r3  = base[3 * kHW];
  float obj = base[4 * kHW];

  float best = base[5 * kHW];
  int bi = 0;
#pragma unroll 8
  for (int c = 1; c < kClasses; ++c) {
    float v = base[(5 + c) * kHW];
    if (v > best) { best = v; bi = c; }   // strict '>' keeps first max (jnp.argmax)
  }

  float x  = (r0 + (float)ii) / (float)kW;
  float y  = (r1 + (float)jj) / (float)kH;
  float aw = anchors[2 * a]     / (float)kW;
  float ah = anchors[2 * a + 1] / (float)kH;
  float wv = expf(r2) * aw;
  float hv = expf(r3) * ah;

  float* o = det + (size_t)t * 7;
  o[0] = obj; o[1] = x; o[2] = y; o[3] = wv; o[4] = hv;
  o[5] = (float)bi; o[6] = best;
}

// ---------------- kernel 2: stable sort + greedy NMS (one block per batch) ---
__global__ __launch_bounds__(1024) void yolo_nms_kernel(
    const float* __restrict__ det,
    float* __restrict__ detOut,     // [kB, kN, 7]
    float* __restrict__ keepOut) {  // [kB, kN]
  const int b   = blockIdx.x;
  const int tid = threadIdx.x;

  __shared__ unsigned long long skey[kNPad];  // sort keys; reused as area after sort
  __shared__ float sl[kNPad], sr[kNPad], su[kNPad], sd[kNPad], sp[kNPad];

  const float* detB = det + (size_t)b * kN * 7;
  const unsigned long long detBase = (unsigned long long)detB;

  // ---- phase 0: async global->LDS: obj -> sl, prob -> sr ----
  for (int n = tid; n < kN; n += 1024) {
    unsigned rowOff = (unsigned)(n * 7 * 4);
    asyncLoadB32(ldsAddr(&sl[n]), rowOff + 0u,  detBase);  // obj  (col 0)
    asyncLoadB32(ldsAddr(&sr[n]), rowOff + 24u, detBase);  // prob (col 6)
  }
  waitAsync0();
  __syncthreads();

  // ---- phase 1: build 64-bit keys: descending prob_eff, stable by index ----
  for (int s = tid; s < kNPad; s += 1024) {
    unsigned long long k;
    if (s < kN) {
      float obj  = sl[s];
      float prob = sr[s];
      float pe   = (obj > kObjTh) ? prob : -1.0f;
      unsigned u = __float_as_uint(pe);
      u = (u & 0x80000000u) ? ~u : (u | 0x80000000u);   // order-preserving map
      k = ((unsigned long long)(~u) << 32) | (unsigned)s;
    } else {
      k = ~0ull;                                        // pads sort last
    }
    skey[s] = k;
  }

  // ---- phase 2: bitonic sort, ascending (=> descending prob_eff) ----
  for (int k2 = 2; k2 <= kNPad; k2 <<= 1) {
    for (int j2 = k2 >> 1; j2 > 0; j2 >>= 1) {
      __syncthreads();
      for (int s = tid; s < kNPad; s += 1024) {
        int ixj = s ^ j2;
        if (ixj > s) {
          bool up = ((s & k2) == 0);
          unsigned long long ka = skey[s], kb = skey[ixj];
          if ((ka > kb) == up) { skey[s] = kb; skey[ixj] = ka; }
        }
      }
    }
  }
  __syncthreads();

  // ---- phase 3: gather sorted rows, emit cols 0..5, fill SoA boxes ----
  unsigned long long k0 = skey[tid];
  unsigned long long k1 = skey[tid + 1024];
  __syncthreads();                      // all key reads done before aliasing
  float* sa = (float*)skey;             // reuse key LDS as area array

  for (int pass = 0; pass < 2; ++pass) {
    int s = tid + pass * 1024;
    unsigned long long kk = pass ? k1 : k0;
    if (s < kN) {
      int ord = (int)(kk & 2047u);
      const float* row = detB + (size_t)ord * 7;
      float obj = row[0], x = row[1], y = row[2];
      float wv = row[3], hv = row[4], cls = row[5], prob = row[6];
      float pe = (obj > kObjTh) ? prob : -1.0f;

      float* o = detOut + ((size_t)b * kN + s) * 7;
      o[0] = obj; o[1] = x; o[2] = y; o[3] = wv; o[4] = hv; o[5] = cls;

      sl[s] = x - 0.5f * wv;  sr[s] = x + 0.5f * wv;
      su[s] = y - 0.5f * hv;  sd[s] = y + 0.5f * hv;
      sa[s] = wv * hv;        // exact union term as in reference
      sp[s] = pe;
    }
  }
  __syncthreads();

  // ---- phase 4: greedy suppression scan ----
  const int j0 = tid, j1 = tid + 1024;
  float l0 = 0, r0 = 0, u0 = 0, d0 = 0, a0 = 0;
  float l1 = 0, r1 = 0, u1 = 0, d1 = 0, a1 = 0;
  if (j0 < kN) { l0 = sl[j0]; r0 = sr[j0]; u0 = su[j0]; d0 = sd[j0]; a0 = sa[j0]; }
  if (j1 < kN) { l1 = sl[j1]; r1 = sr[j1]; u1 = su[j1]; d1 = sd[j1]; a1 = sa[j1]; }

  for (int i = 0; i < kN - 1; ++i) {
    float pi = sp[i];                   // uniform across the block between barriers
    if (pi > 0.0f) {
      float li = sl[i], ri = sr[i], ui = su[i], di = sd[i], ai = sa[i];
      if (j0 > i && j0 < kN) {
        float iw = fmaxf(fminf(ri, r0) - fmaxf(li, l0), 0.0f);
        float ih = fmaxf(fminf(di, d0) - fmaxf(ui, u0), 0.0f);
        float inter = iw * ih;
        float iou = inter / (ai + a0 - inter);
        if (iou >= kNmsTh) sp[j0] *= 0.0f;   // preserves -0.0 semantics
      }
      if (j1 > i && j1 < kN) {
        float iw = fmaxf(fminf(ri, r1) - fmaxf(li, l1), 0.0f);
        float ih = fmaxf(fminf(di, d1) - fmaxf(ui, u1), 0.0f);
        float inter = iw * ih;
        float iou = inter / (ai + a1 - inter);
        if (iou >= kNmsTh) sp[j1] *= 0.0f;
      }
      __syncthreads();                  // uniform branch -> safe barrier
    }
    // dead i: no writes happened since last barrier -> no barrier needed
  }
  __syncthreads();

  // ---- phase 5: final prob column + keep flags ----
  for (int s = tid; s < kN; s += 1024) {
    float pf = sp[s];                   // final by construction (only j>i written)
    detOut[((size_t)b * kN + s) * 7 + 6] = pf;
    keepOut[(size_t)b * kN + s] = (pf > 0.0f && s < kN - 1) ? 1.0f : 0.0f;
  }
}

// ---------------- host launch ----------------
extern "C" void kernel_launch(void* const* d_in, const int* in_sizes, int n_in,
                              void* d_out, int out_size, void* d_ws, size_t ws_size,
                              hipStream_t stream) {
  const float* pred    = (const float*)d_in[0];   // (32, 425, 19, 19) f32
  const float* anchors = (const float*)d_in[1];   // (10,) f32

  float* det     = (float*)d_ws;                  // [kB, kN, 7] scratch
  float* detOut  = (float*)d_out;                 // [kB, kN, 7]
  float* keepOut = detOut + (size_t)kB * kN * 7;  // [kB, kN]

  int total = kB * kN;
  yolo_decode_kernel<<<(total + 255) / 256, 256, 0, stream>>>(pred, anchors, det);
  yolo_nms_kernel<<<kB, 1024, 0, stream>>>(det, detOut, keepOut);
}